// SAGENodeClassifier_15814069584246
// MI455X (gfx1250) — compile-verified
//
#include <hip/hip_runtime.h>

// ---------------------------------------------------------------------------
// GraphSAGE node classifier for MI455X (gfx1250, wave32).
//   conv: out = relu(mean_neighbors(x) @ Wl^T + bl + x @ Wr^T)
//   head: h2 @ Wc^T + bc
// Scatter phase is L2-atomic bound (~1 GB irregular traffic; the 25.6 MB
// feature matrix is L2-resident on the 192 MB L2). Dense linears run on the
// matrix pipe via V_WMMA_F32_16X16X4_F32 at full fp32 precision, with the
// A-operand staged through LDS (coalesced float4 stage-in, ds_load_b64
// fragment reads) and a pure 16-deep WMMA accumulation chain.
// ---------------------------------------------------------------------------

typedef float v2f __attribute__((ext_vector_type(2)));
typedef float v8f __attribute__((ext_vector_type(8)));

#define HID 64

// --- edge scatter: agg[dst] += x[src], 4 features per thread -----------------
__global__ __launch_bounds__(256)
void scatter_feat_kernel(const float* __restrict__ x,
                         const int* __restrict__ src,
                         const int* __restrict__ dst,
                         float* __restrict__ agg,
                         long long total /* = E*16 */)
{
    long long tid = (long long)blockIdx.x * blockDim.x + threadIdx.x;
    if (tid >= total) return;
    int e = (int)(tid >> 4);
    int f = (int)(tid & 15) * 4;            // 16-byte aligned feature quad
    int s = src[e];
    int d = dst[e];
    const float4 v = *(const float4*)(x + (size_t)s * HID + f);
    float* a = agg + (size_t)d * HID + f;
    atomicAdd(a + 0, v.x);
    atomicAdd(a + 1, v.y);
    atomicAdd(a + 2, v.z);
    atomicAdd(a + 3, v.w);
}

__global__ __launch_bounds__(256)
void count_kernel(const int* __restrict__ dst, float* __restrict__ cnt, int E)
{
    int e = blockIdx.x * blockDim.x + threadIdx.x;
    if (e < E) atomicAdd(&cnt[dst[e]], 1.0f);
}

__global__ __launch_bounds__(256)
void invcnt_kernel(float* __restrict__ cnt, int N)
{
    int n = blockIdx.x * blockDim.x + threadIdx.x;
    if (n < N) cnt[n] = 1.0f / fmaxf(cnt[n], 1.0f);
}

// --- SAGE linear layer via fp32 WMMA ----------------------------------------
// Block = 128 threads (4 waves) -> one 16-node tile, each wave one 16-col tile.
// A fragment (16x4):  lane L -> row M=L&15,  K = 4s + 2*(L>>4) + {0,1}
// B fragment (4x16):  lane L -> col N=L&15,  same K indices (row-striped)
// C/D (16x16, 8 VGPR): VGPR r -> row r + 8*(L>>4), col L&15
__global__ __launch_bounds__(128)
void sage_linear_wmma(const float* __restrict__ agg,
                      const float* __restrict__ inv,
                      const float* __restrict__ xin,
                      const float* __restrict__ Wl,   // [64,64] row-major (out,in)
                      const float* __restrict__ bl,   // [64]
                      const float* __restrict__ Wr,   // [64,64]
                      float* __restrict__ out,        // [N,64]
                      int N)
{
    __shared__ float sA[16 * HID];   // inv-degree scaled neighbor sums
    __shared__ float sX[16 * HID];   // self features

    const int tid = threadIdx.x;

    // ---- cooperative coalesced stage-in: 16 rows x 64 floats per buffer ----
    {
        const int row = tid >> 3;              // 0..15
        const int off = (tid & 7) * 8;         // 0,8,...,56
        int node = blockIdx.x * 16 + row;
        if (node > N - 1) node = N - 1;        // clamp: loads unconditional
        const float ic = inv[node];
        const float4* gA = (const float4*)(agg + (size_t)node * HID + off);
        const float4* gX = (const float4*)(xin + (size_t)node * HID + off);
        const float4 a0 = gA[0], a1 = gA[1];
        const float4 x0 = gX[0], x1 = gX[1];
        float4* dA = (float4*)(sA + row * HID + off);
        float4* dX = (float4*)(sX + row * HID + off);
        dA[0] = make_float4(a0.x * ic, a0.y * ic, a0.z * ic, a0.w * ic);
        dA[1] = make_float4(a1.x * ic, a1.y * ic, a1.z * ic, a1.w * ic);
        dX[0] = x0;
        dX[1] = x1;
    }
    __syncthreads();

    const int lane = tid & 31;
    const int wave = tid >> 5;                 // output column tile 0..3
    const int m    = lane & 15;
    const int hi   = lane >> 4;                // K half selector
    const int j    = wave * 16 + m;            // output column
    const float* wlRow = Wl + j * HID;         // B(k,j) = W[j,k]
    const float* wrRow = Wr + j * HID;

    v2f aF[16], bF[16];
    v8f acc = {};

    // GEMM 1: mean(neighbors) @ Wl^T  (mean pre-scaled at stage-in)
    #pragma unroll
    for (int s = 0; s < 16; ++s) {
        const int k = s * 4 + hi * 2;
        aF[s] = *(const v2f*)(sA + m * HID + k);
        bF[s] = *(const v2f*)(wlRow + k);
    }
    #pragma unroll
    for (int s = 0; s < 16; ++s)
        acc = __builtin_amdgcn_wmma_f32_16x16x4_f32(
            false, aF[s], false, bF[s], (short)0, acc, false, false);

    // GEMM 2: x @ Wr^T, accumulate into same tile
    #pragma unroll
    for (int s = 0; s < 16; ++s) {
        const int k = s * 4 + hi * 2;
        aF[s] = *(const v2f*)(sX + m * HID + k);
        bF[s] = *(const v2f*)(wrRow + k);
    }
    #pragma unroll
    for (int s = 0; s < 16; ++s)
        acc = __builtin_amdgcn_wmma_f32_16x16x4_f32(
            false, aF[s], false, bF[s], (short)0, acc, false, false);

    const float bias = bl[j];
    #pragma unroll
    for (int r = 0; r < 8; ++r) {
        const int row = r + 8 * hi;
        const int node = blockIdx.x * 16 + row;
        if (node < N)
            out[(size_t)node * HID + j] = fmaxf(acc[r] + bias, 0.0f);
    }
}

// --- classifier: h2 @ Wc^T + bc  (40 classes -> 3 col tiles, mask j>=40) ----
__global__ __launch_bounds__(96)
void classifier_wmma(const float* __restrict__ h,
                     const float* __restrict__ Wc,   // [40,64]
                     const float* __restrict__ bc,   // [40]
                     float* __restrict__ out,        // [N,40]
                     int N, int C)
{
    __shared__ float sH[16 * HID];

    const int tid = threadIdx.x;                // 0..95 (3 waves)
    {
        // 96 threads stage 1024 floats: rows 0..15, ~10.7 floats each ->
        // use up to 3 passes of float4 per thread over a flat index space.
        const int quads = (16 * HID) / 4;       // 256 float4s
        for (int q = tid; q < quads; q += 96) {
            const int row = q >> 4;             // 16 quads per row
            const int off = (q & 15) * 4;
            int node = blockIdx.x * 16 + row;
            if (node > N - 1) node = N - 1;
            *(float4*)(sH + row * HID + off) =
                *(const float4*)(h + (size_t)node * HID + off);
        }
    }
    __syncthreads();

    const int lane = tid & 31;
    const int wave = tid >> 5;                  // column tile 0..2
    const int m    = lane & 15;
    const int hi   = lane >> 4;
    const int j    = wave * 16 + m;
    const bool okJ = j < C;
    const int jc   = okJ ? j : (C - 1);         // clamp: cols are independent
    const float* wRow = Wc + jc * HID;

    v2f aF[16], bF[16];
    v8f acc = {};
    #pragma unroll
    for (int s = 0; s < 16; ++s) {
        const int k = s * 4 + hi * 2;
        aF[s] = *(const v2f*)(sH + m * HID + k);
        bF[s] = *(const v2f*)(wRow + k);
    }
    #pragma unroll
    for (int s = 0; s < 16; ++s)
        acc = __builtin_amdgcn_wmma_f32_16x16x4_f32(
            false, aF[s], false, bF[s], (short)0, acc, false, false);

    const float bias = bc[jc];
    #pragma unroll
    for (int r = 0; r < 8; ++r) {
        const int row = r + 8 * hi;
        const int node = blockIdx.x * 16 + row;
        if (node < N && okJ)
            out[(size_t)node * C + j] = acc[r] + bias;
    }
}

// ---------------------------------------------------------------------------
extern "C" void kernel_launch(void* const* d_in, const int* in_sizes, int n_in,
                              void* d_out, int out_size, void* d_ws, size_t ws_size,
                              hipStream_t stream)
{
    const float* x   = (const float*)d_in[0];
    const int*   ei  = (const int*)d_in[1];
    const float* W1l = (const float*)d_in[2];
    const float* b1l = (const float*)d_in[3];
    const float* W1r = (const float*)d_in[4];
    const float* W2l = (const float*)d_in[5];
    const float* b2l = (const float*)d_in[6];
    const float* W2r = (const float*)d_in[7];
    const float* Wc  = (const float*)d_in[8];
    const float* bc  = (const float*)d_in[9];
    float* out = (float*)d_out;

    const int N = in_sizes[0] / HID;
    const int E = in_sizes[1] / 2;
    const int C = in_sizes[9];               // 40 classes
    const int* src = ei;
    const int* dst = ei + E;

    // workspace: agg[N*64] | cnt[N] | h1[N*64] | h2[N*64]  (~77 MB)
    float* agg = (float*)d_ws;
    float* cnt = agg + (size_t)N * HID;
    float* h1  = cnt + N;
    float* h2  = h1 + (size_t)N * HID;

    const long long totalEF = (long long)E * (HID / 4);   // 4 feats per thread
    const unsigned scatterBlocks = (unsigned)((totalEF + 255) / 256);
    const int nodeTiles = (N + 15) / 16;

    // ---- layer 1 ----
    hipMemsetAsync(agg, 0, ((size_t)N * HID + N) * sizeof(float), stream); // agg+cnt
    scatter_feat_kernel<<<scatterBlocks, 256, 0, stream>>>(x, src, dst, agg, totalEF);
    count_kernel<<<(E + 255) / 256, 256, 0, stream>>>(dst, cnt, E);
    invcnt_kernel<<<(N + 255) / 256, 256, 0, stream>>>(cnt, N);   // cnt -> 1/max(deg,1)
    sage_linear_wmma<<<nodeTiles, 128, 0, stream>>>(agg, cnt, x, W1l, b1l, W1r, h1, N);

    // ---- layer 2 (same graph: reuse inverse degree) ----
    hipMemsetAsync(agg, 0, (size_t)N * HID * sizeof(float), stream);
    scatter_feat_kernel<<<scatterBlocks, 256, 0, stream>>>(h1, src, dst, agg, totalEF);
    sage_linear_wmma<<<nodeTiles, 128, 0, stream>>>(agg, cnt, h1, W2l, b2l, W2r, h2, N);

    // ---- classifier head ----
    classifier_wmma<<<nodeTiles, 96, 0, stream>>>(h2, Wc, bc, out, N, C);
}